// FastRCNNOutputLayers_34565896798855
// MI455X (gfx1250) — compile-verified
//
#include <hip/hip_runtime.h>
#include <hip/hip_bf16.h>

// FastRCNN output layer post-processing for gfx1250 (MI455X).
//  Kernel A: per-class (80 WGs) TDM-staged load -> bitonic sort -> greedy NMS.
//  Kernel B: global top-100 via persistent per-stripe maxima (tournament
//            extraction: only the winner's owner thread rescans its stripe).
// CDNA5-specific paths: tensor_load_to_lds (TDM), s_wait_tensorcnt,
//                       LDS ds atomics, global_prefetch_b8.

#define SCORE_THRESH 0.05f
#define NMS_THRESH   0.5f
#define TOPK   100
#define RBOX   1000
#define NCLS   80
#define NSORT  1024
#define NEGV   (-1.0f)
#define FNEGMAX (-3.402823466e38f)
#define TPB_B  256

typedef unsigned int u32x4 __attribute__((ext_vector_type(4)));
typedef int          i32x4 __attribute__((ext_vector_type(4)));
typedef int          i32x8 __attribute__((ext_vector_type(8)));

// Low 32 bits of a generic pointer to LDS == byte offset within the
// workgroup's LDS allocation (aperture scheme: addr[63:32]=SHARED_BASE).
__device__ __forceinline__ unsigned lds_byte_offset(const void* p) {
  return (unsigned)(unsigned long long)p;
}

// Issue one TDM 2D-tile load: tile_d1 rows of tile_d0 fp32 elements, row
// stride = stride0_elems, packed contiguously into LDS at lds_addr.
// D# layout per CDNA5 ISA ch.8 (group0: count/lds/global/type, group1:
// data_size/dims/tile/strides). Groups 2/3 zero (<=2D tensor).
__device__ __forceinline__ void tdm_load_2d_f32(unsigned lds_addr, const float* gptr,
                                                unsigned tensor_d0, unsigned tensor_d1,
                                                unsigned tile_d0, unsigned tile_d1,
                                                unsigned stride0_elems) {
  unsigned long long ga = (unsigned long long)gptr;
  u32x4 g0 = {0u, 0u, 0u, 0u};
  g0.x = 1u;                                            // count=1, user descriptor
  g0.y = lds_addr;                                      // lds_addr[31:0]
  g0.z = (unsigned)ga;                                  // global_addr[31:0]
  g0.w = ((unsigned)(ga >> 32) & 0x1FFFFFFu)            // global_addr[56:32]
       | (2u << 30);                                    // type=2 ("image")
  i32x8 g1 = {0, 0, 0, 0, 0, 0, 0, 0};
  g1[0] = (int)(2u << 16);                              // data_size=2 (4 bytes), wg_mask=0
  g1[1] = (int)((tensor_d0 & 0xFFFFu) << 16);           // tensor_dim0[15:0] @ bits 63:48
  g1[2] = (int)(((tensor_d0 >> 16) & 0xFFFFu)           // tensor_dim0[31:16]
              | ((tensor_d1 & 0xFFFFu) << 16));         // tensor_dim1[15:0]
  g1[3] = (int)(((tensor_d1 >> 16) & 0xFFFFu)           // tensor_dim1[31:16]
              | ((tile_d0 & 0xFFFFu) << 16));           // tile_dim0
  g1[4] = (int)(tile_d1 & 0xFFFFu);                     // tile_dim1 (tile_dim2=0)
  g1[5] = (int)stride0_elems;                           // tensor_dim0_stride[31:0]
  g1[6] = 0;                                            // stride0 hi / stride1 lo (unused, 2D)
  g1[7] = 0;
  i32x4 g2 = {0, 0, 0, 0};
  i32x4 g3 = {0, 0, 0, 0};
#if defined(__clang_major__) && (__clang_major__ >= 23)
  i32x8 g4 = {0, 0, 0, 0, 0, 0, 0, 0};
  __builtin_amdgcn_tensor_load_to_lds(g0, g1, g2, g3, g4, 0);
#else
  __builtin_amdgcn_tensor_load_to_lds(g0, g1, g2, g3, 0);
#endif
}

__global__ __launch_bounds__(256)
void nms_per_class_kernel(const float* __restrict__ boxes,
                          const float* __restrict__ scores,
                          const int* __restrict__ p_imh,
                          const int* __restrict__ p_imw,
                          float* __restrict__ ws_scores,
                          float* __restrict__ ws_boxes) {
  __shared__ float    rawbox[RBOX * 4];      // TDM dest: this class's boxes
  __shared__ float    rawscore[RBOX];        // TDM dest: this class's scores
  __shared__ float    skey[NSORT];           // sort keys (scores), later masked
  __shared__ int      sidx[NSORT];           // sort payload (original row)
  __shared__ float    sbox[NSORT * 4];       // clipped boxes in sorted order
  __shared__ unsigned activew[NSORT / 32];   // NMS active bitmask

  const int tid = threadIdx.x;
  const int cls = blockIdx.x;

  // ---- Stage inputs via Tensor Data Mover (wave 0 issues, EXEC-independent) ----
  if (tid < 32) {
    // boxes: row r at boxes[r*320 + cls*4 .. +3]  -> 1000 x 4 f32 tile, stride 320
    tdm_load_2d_f32(lds_byte_offset(rawbox), boxes + cls * 4,
                    /*tensor_d0=*/NCLS * 4, /*tensor_d1=*/RBOX,
                    /*tile_d0=*/4, /*tile_d1=*/RBOX, /*stride0=*/NCLS * 4);
    // scores: row r at scores[r*81 + cls]         -> 1000 x 1 f32 tile, stride 81
    tdm_load_2d_f32(lds_byte_offset(rawscore), scores + cls,
                    /*tensor_d0=*/NCLS + 1, /*tensor_d1=*/RBOX,
                    /*tile_d0=*/1, /*tile_d1=*/RBOX, /*stride0=*/NCLS + 1);
    __builtin_amdgcn_s_wait_tensorcnt(0);
  }
  if (tid < NSORT / 32) activew[tid] = 0xFFFFFFFFu;
  __syncthreads();

  // ---- Init sort arrays (pad 1000..1023 with -inf keys) ----
  for (int i = tid; i < NSORT; i += 256) {
    skey[i] = (i < RBOX) ? rawscore[i] : FNEGMAX;
    sidx[i] = i;
  }

  // ---- Bitonic sort, descending score, stable (ascending index) on ties ----
  for (unsigned k = 2; k <= NSORT; k <<= 1) {
    for (unsigned j = k >> 1; j > 0; j >>= 1) {
      __syncthreads();
      for (unsigned i = tid; i < NSORT; i += 256) {
        unsigned ixj = i ^ j;
        if (ixj > i) {
          float ka = skey[i], kb = skey[ixj];
          int   ia = sidx[i], ib = sidx[ixj];
          bool aLess = (ka < kb) || (ka == kb && ia > ib);  // total order
          bool desc  = ((i & k) == 0);
          if (aLess == desc) {
            skey[i] = kb; skey[ixj] = ka;
            sidx[i] = ib; sidx[ixj] = ia;
          }
        }
      }
    }
  }
  __syncthreads();

  // ---- Gather boxes in sorted order, clip to image ----
  const float W = (float)p_imw[0];
  const float H = (float)p_imh[0];
  for (int i = tid; i < NSORT; i += 256) {
    int id = sidx[i];
    float x0 = 0.f, y0 = 0.f, x1 = 0.f, y1 = 0.f;
    if (id < RBOX) {
      x0 = fminf(fmaxf(rawbox[id * 4 + 0], 0.f), W);
      y0 = fminf(fmaxf(rawbox[id * 4 + 1], 0.f), H);
      x1 = fminf(fmaxf(rawbox[id * 4 + 2], 0.f), W);
      y1 = fminf(fmaxf(rawbox[id * 4 + 3], 0.f), H);
    }
    sbox[i * 4 + 0] = x0; sbox[i * 4 + 1] = y0;
    sbox[i * 4 + 2] = x1; sbox[i * 4 + 3] = y1;
  }
  __syncthreads();

  // ---- Greedy NMS: sequential over kept candidates, parallel suppression ----
  for (int i = 0; i < RBOX; ++i) {
    float si = skey[i];
    if (!(si > SCORE_THRESH)) {
      // sorted descending: everything from here is invalid -> score = -1
      for (int j = i + tid; j < RBOX; j += 256) skey[j] = NEGV;
      break;
    }
    bool act = (activew[i >> 5] >> (i & 31)) & 1u;
    if (act) {  // kept: suppress overlapping later boxes in parallel
      float bx0 = sbox[i * 4 + 0], by0 = sbox[i * 4 + 1];
      float bx1 = sbox[i * 4 + 2], by1 = sbox[i * 4 + 3];
      float ai = (bx1 - bx0) * (by1 - by0);
      for (int j = i + 1 + tid; j < RBOX; j += 256) {
        float cx0 = sbox[j * 4 + 0], cy0 = sbox[j * 4 + 1];
        float cx1 = sbox[j * 4 + 2], cy1 = sbox[j * 4 + 3];
        float iw = fmaxf(fminf(bx1, cx1) - fmaxf(bx0, cx0), 0.f);
        float ih = fmaxf(fminf(by1, cy1) - fmaxf(by0, cy0), 0.f);
        float inter = iw * ih;
        float aj = (cx1 - cx0) * (cy1 - cy0);
        float uni = fmaxf(ai + aj - inter, 1e-9f);
        if (inter > NMS_THRESH * uni) {          // iou > 0.5
          atomicAnd(&activew[j >> 5], ~(1u << (j & 31)));   // ds_and_b32
        }
      }
    } else {
      if (tid == 0) skey[i] = NEGV;              // suppressed -> -1
    }
    __syncthreads();
  }
  __syncthreads();

  // ---- Emit per-class results to workspace (flat index = cls*R + r) ----
  float* oS = ws_scores + cls * RBOX;
  float* oB = ws_boxes + cls * RBOX * 4;
  for (int i = tid; i < RBOX; i += 256)     oS[i] = skey[i];
  for (int i = tid; i < RBOX * 4; i += 256) oB[i] = sbox[i];
}

__global__ __launch_bounds__(TPB_B)
void topk_kernel(float* __restrict__ ws_scores,
                 const float* __restrict__ ws_boxes,
                 float* __restrict__ out) {
  __shared__ float pb[TPB_B];  __shared__ int pbi[TPB_B];   // persistent stripe bests
  __shared__ float rs[TPB_B];  __shared__ int ri[TPB_B];    // reduction workspace
  const int tid = threadIdx.x;
  const int N = NCLS * RBOX;

  // Initial per-thread stripe maximum. Strict '>' keeps the earliest index
  // within a stripe (stripe indices are monotonically increasing).
  float best = FNEGMAX; int besti = N;
  for (int m = tid; m < N; m += TPB_B) {
    __builtin_prefetch(&ws_scores[m + 8 * TPB_B], 0, 1);   // global_prefetch_b8
    float v = ws_scores[m];
    if (v > best) { best = v; besti = m; }
  }
  pb[tid] = best; pbi[tid] = besti;
  __syncthreads();

  for (int k = 0; k < TOPK; ++k) {
    // Reduce the 256 stripe bests: higher score wins; ties -> lower flat
    // index (matches lax.top_k ordering).
    rs[tid] = pb[tid]; ri[tid] = pbi[tid];
    __syncthreads();
    for (int off = TPB_B / 2; off > 0; off >>= 1) {
      if (tid < off) {
        float vo = rs[tid + off]; int io = ri[tid + off];
        if (vo > rs[tid] || (vo == rs[tid] && io < ri[tid])) {
          rs[tid] = vo; ri[tid] = io;
        }
      }
      __syncthreads();
    }
    int   m = ri[0];     // uniform across block
    float v = rs[0];
    if (tid == 0) {
      out[k * 5 + 0] = ws_boxes[m * 4 + 0];
      out[k * 5 + 1] = ws_boxes[m * 4 + 1];
      out[k * 5 + 2] = ws_boxes[m * 4 + 2];
      out[k * 5 + 3] = ws_boxes[m * 4 + 3];
      out[k * 5 + 4] = v;
      out[TOPK * 5 + k] = (float)(m / RBOX);   // class id
    }
    // Only the extracted element's owner stripe changed: that thread
    // invalidates it and rescans its own stripe; all other bests stay valid.
    if (tid == (m % TPB_B)) {
      ws_scores[m] = FNEGMAX;
      float nb = FNEGMAX; int nbi = N;
      for (int mm = tid; mm < N; mm += TPB_B) {
        float vv = ws_scores[mm];
        if (vv > nb) { nb = vv; nbi = mm; }
      }
      pb[tid] = nb; pbi[tid] = nbi;
    }
    __syncthreads();
  }
}

extern "C" void kernel_launch(void* const* d_in, const int* in_sizes, int n_in,
                              void* d_out, int out_size, void* d_ws, size_t ws_size,
                              hipStream_t stream) {
  (void)in_sizes; (void)n_in; (void)out_size; (void)ws_size;
  const float* boxes  = (const float*)d_in[0];   // (1000, 320) f32
  const float* scores = (const float*)d_in[1];   // (1000, 81)  f32
  const int*   imh    = (const int*)d_in[2];     // scalar
  const int*   imw    = (const int*)d_in[3];     // scalar

  float* ws_scores = (float*)d_ws;               // 80,000 f32 (rewritten fully every call)
  float* ws_boxes  = ws_scores + NCLS * RBOX;    // 320,000 f32
  float* out       = (float*)d_out;              // 500 dets + 100 classes

  hipLaunchKernelGGL(nms_per_class_kernel, dim3(NCLS), dim3(256), 0, stream,
                     boxes, scores, imh, imw, ws_scores, ws_boxes);
  hipLaunchKernelGGL(topk_kernel, dim3(1), dim3(TPB_B), 0, stream,
                     ws_scores, ws_boxes, out);
}